// GraphAutoencoder_10264971837865
// MI455X (gfx1250) — compile-verified
//
#include <hip/hip_runtime.h>

typedef __attribute__((ext_vector_type(2))) float v2f;
typedef __attribute__((ext_vector_type(8))) float v8f;

// ---------------- degree / normalization ----------------

__global__ void fill_ones_kernel(float* __restrict__ p, int n) {
  int i = blockIdx.x * blockDim.x + threadIdx.x;
  if (i < n) p[i] = 1.0f;  // self-loop contributes 1 to every node's degree
}

__global__ void deg_accum_kernel(float* __restrict__ deg, const int* __restrict__ dst, int E) {
  int e = blockIdx.x * blockDim.x + threadIdx.x;
  if (e < E) atomicAdd(&deg[dst[e]], 1.0f);
}

__global__ void to_rsqrt_kernel(float* __restrict__ p, int n) {
  int i = blockIdx.x * blockDim.x + threadIdx.x;
  if (i < n) p[i] = rsqrtf(p[i]);  // deg >= 1 always (self-loops)
}

// ---------------- dense transform: out[N,FOUT] = A[N,FIN] @ W[FIN,FOUT] ----------------
// One wave per 16x16 output tile, full-precision f32 WMMA (16x16x4), K chained.
// A frag: lane m = lane%16, VGPR j holds K = 2*(lane/16)+j  -> aligned float2 load.
// B frag: lane n = lane%16, VGPR j holds K = 2*(lane/16)+j.
// C/D  : VGPR r -> row r + 8*(lane/16), col lane%16.

template <int FIN, int FOUT>
__global__ __launch_bounds__(128) void gemm_wmma_kernel(const float* __restrict__ A,
                                                        const float* __restrict__ W,
                                                        float* __restrict__ out) {
  const int wave = threadIdx.x >> 5;
  const int lane = threadIdx.x & 31;
  const int lg = lane >> 4;   // lane group 0/1
  const int lm = lane & 15;
  const int m0 = blockIdx.x * 16;
  const int n0 = wave * 16;

  const float* arow = A + (size_t)(m0 + lm) * FIN + 2 * lg;
  const float* bcol = W + (size_t)(2 * lg) * FOUT + (n0 + lm);

  v8f c = {};
#pragma unroll 4
  for (int k = 0; k < FIN; k += 4) {
    v2f a = *(const v2f*)(arow + k);                 // A[m][k+2lg .. k+2lg+1]
    v2f b;
    b.x = bcol[(size_t)k * FOUT];                    // B[k+2lg  ][n]
    b.y = bcol[(size_t)(k + 1) * FOUT];              // B[k+2lg+1][n]
    c = __builtin_amdgcn_wmma_f32_16x16x4_f32(false, a, false, b, (short)0, c, false, false);
  }

  float* orow = out + (size_t)(m0 + 8 * lg) * FOUT + (n0 + lm);
#pragma unroll
  for (int r = 0; r < 8; ++r) orow[(size_t)r * FOUT] = c[r];
}

// ---------------- aggregation ----------------
// out[i] = h[i] * dinv[i]^2   (self-loop message; also overwrites poison)
template <int F>
__global__ void selfloop_init_kernel(const float* __restrict__ h, const float* __restrict__ dinv,
                                     float* __restrict__ out, int n) {
  long t = (long)blockIdx.x * blockDim.x + threadIdx.x;
  if (t >= (long)n * F) return;
  int i = (int)(t / F);
  float di = dinv[i];
  out[t] = h[t] * (di * di);
}

// out[dst] += h[src] * dinv[src]*dinv[dst], one thread per (edge, feature)
template <int F>
__global__ void edge_scatter_kernel(const float* __restrict__ h, const float* __restrict__ dinv,
                                    const int* __restrict__ src, const int* __restrict__ dst,
                                    float* __restrict__ out, int E) {
  long t = (long)blockIdx.x * blockDim.x + threadIdx.x;
  if (t >= (long)E * F) return;
  int e = (int)(t / F);
  int f = (int)(t % F);
  int s = src[e];
  int d = dst[e];
  float nrm = dinv[s] * dinv[d];
  atomicAdd(&out[(size_t)d * F + f], h[(size_t)s * F + f] * nrm);
}

template <int F, bool RELU>
__global__ void bias_act_kernel(float* __restrict__ a, const float* __restrict__ b, int n) {
  long t = (long)blockIdx.x * blockDim.x + threadIdx.x;
  if (t >= (long)n * F) return;
  float v = a[t] + b[(int)(t % F)];
  a[t] = RELU ? fmaxf(v, 0.0f) : v;
}

// ---------------- launch ----------------

extern "C" void kernel_launch(void* const* d_in, const int* in_sizes, int n_in,
                              void* d_out, int out_size, void* d_ws, size_t ws_size,
                              hipStream_t stream) {
  const float* x  = (const float*)d_in[0];   // [N,128]
  const int*   ei = (const int*)d_in[1];     // [2,E]
  const float* W1 = (const float*)d_in[2];   // [128,64]
  const float* b1 = (const float*)d_in[3];   // [64]
  const float* W2 = (const float*)d_in[4];   // [64,32]
  const float* b2 = (const float*)d_in[5];   // [32]
  float* out = (float*)d_out;                // [N,32]

  constexpr int FIN = 128, FH = 64, FOUT = 32;
  const int N = in_sizes[0] / FIN;   // 100000 (divisible by 16)
  const int E = in_sizes[1] / 2;     // 1600000
  const int* src = ei;
  const int* dst = ei + E;

  // carve scratch from d_ws
  char* ws = (char*)d_ws;
  size_t off = 0;
  auto carve = [&](size_t bytes) -> float* {
    float* p = (float*)(ws + off);
    off = (off + bytes + 255) & ~(size_t)255;
    return p;
  };
  float* dinv = carve((size_t)N * sizeof(float));         // deg, then rsqrt in-place
  float* h1   = carve((size_t)N * FH * sizeof(float));    // x @ W1
  float* a1   = carve((size_t)N * FH * sizeof(float));    // layer-1 aggregate (then ReLU'd)
  float* z2   = carve((size_t)N * FOUT * sizeof(float));  // a1 @ W2
  (void)ws_size; (void)n_in; (void)out_size;

  const int T = 256;
  auto nb = [](long total, int t) { return (int)((total + t - 1) / t); };

  // degree -> dinv (shared by both layers)
  fill_ones_kernel<<<nb(N, T), T, 0, stream>>>(dinv, N);
  deg_accum_kernel<<<nb(E, T), T, 0, stream>>>(dinv, dst, E);
  to_rsqrt_kernel<<<nb(N, T), T, 0, stream>>>(dinv, N);

  // layer 1: transform, aggregate, bias+ReLU
  gemm_wmma_kernel<FIN, FH><<<N / 16, (FH / 16) * 32, 0, stream>>>(x, W1, h1);
  selfloop_init_kernel<FH><<<nb((long)N * FH, T), T, 0, stream>>>(h1, dinv, a1, N);
  edge_scatter_kernel<FH><<<nb((long)E * FH, T), T, 0, stream>>>(h1, dinv, src, dst, a1, E);
  bias_act_kernel<FH, true><<<nb((long)N * FH, T), T, 0, stream>>>(a1, b1, N);

  // layer 2: transform, aggregate, bias
  gemm_wmma_kernel<FH, FOUT><<<N / 16, (FOUT / 16) * 32, 0, stream>>>(a1, W2, z2);
  selfloop_init_kernel<FOUT><<<nb((long)N * FOUT, T), T, 0, stream>>>(z2, dinv, out, N);
  edge_scatter_kernel<FOUT><<<nb((long)E * FOUT, T), T, 0, stream>>>(z2, dinv, src, dst, out, E);
  bias_act_kernel<FOUT, false><<<nb((long)N * FOUT, T), T, 0, stream>>>(out, b2, N);
}